// SoftplusmaxInfoNCE_22265110462983
// MI455X (gfx1250) — compile-verified
//
#include <hip/hip_runtime.h>

typedef __attribute__((ext_vector_type(2))) float v2f;
typedef __attribute__((ext_vector_type(8))) float v8f;

#define B_ROWS 8192
#define N_ROWS 16384
#define DIMK   128
#define NTILES (N_ROWS / 16)   // 1024 column tiles
#define KSTEPS (DIMK / 4)      // 32 wmma f32 16x16x4 steps

__global__ void zero_out_kernel(float* out) { out[0] = 0.0f; }

__global__ __launch_bounds__(256)
void infonce_wmma_kernel(const float* __restrict__ za,
                         const float* __restrict__ zb,
                         float* __restrict__ out) {
  const int lane  = threadIdx.x & 31;
  const int wave  = threadIdx.x >> 5;
  const int l16   = lane & 15;     // N index within tile / M row for A loads
  const int lhalf = lane >> 4;     // which K-pair half this lane holds
  const int koff  = lhalf * 2;

  const int mt = blockIdx.x * 8 + wave;   // M tile owned by this wave
  const int m0 = mt * 16;

  // ---- Preload A tile: rows m0..m0+15, full K=128 ----
  // f32 16x4 A layout: lane l holds A[M=l%16][K=4k+2*(l/16) + {0,1}] in a float2
  const int arow = m0 + l16;
  const float* abase = (arow < B_ROWS) ? (za + (size_t)arow * DIMK)
                                       : (zb + (size_t)(arow - B_ROWS) * DIMK);
  v2f areg[KSTEPS];
#pragma unroll
  for (int k = 0; k < KSTEPS; ++k)
    areg[k] = *(const v2f*)(abase + k * 4 + koff);

  float rsum[8], spos[8];
  int   grow[8];
#pragma unroll
  for (int v = 0; v < 8; ++v) {
    rsum[v] = 0.0f;
    spos[v] = 0.0f;
    grow[v] = m0 + v + 8 * lhalf;   // C/D layout: VGPR v -> row v (lanes 0-15), v+8 (lanes 16-31)
  }

  // ---- Sweep all 1024 column tiles ----
  for (int nt = 0; nt < NTILES; ++nt) {
    const int n0   = nt * 16;
    const int brow = n0 + l16;      // B is 4x16 KxN: same per-lane load pattern as A
    const float* bbase = (brow < B_ROWS) ? (za + (size_t)brow * DIMK)
                                         : (zb + (size_t)(brow - B_ROWS) * DIMK);

    v8f c = {0.f, 0.f, 0.f, 0.f, 0.f, 0.f, 0.f, 0.f};
#pragma unroll
    for (int k = 0; k < KSTEPS; ++k) {
      v2f b = *(const v2f*)(bbase + k * 4 + koff);
      c = __builtin_amdgcn_wmma_f32_16x16x4_f32(
              /*neg_a=*/false, areg[k], /*neg_b=*/false, b,
              /*c_mod=*/(short)0, c, /*reuse_a=*/false, /*reuse_b=*/false);
    }

    const int gc = n0 + l16;        // global column of every element in this lane
#pragma unroll
    for (int v = 0; v < 8; ++v) {
      const float x  = 2.0f * c[v];                 // sim = dot / TAU, TAU = 0.5
      // numerically stable softplus: max(x,0) + log1p(exp(-|x|))
      const float sp = fmaxf(x, 0.0f) + __logf(1.0f + __expf(-fabsf(x)));
      const int   gr = grow[v];
      rsum[v] += (gc == gr) ? 0.0f : sp;                         // zero diagonal
      spos[v] += (gc == ((gr + B_ROWS) & (N_ROWS - 1))) ? sp : 0.0f;  // positive pair
    }
  }

  // ---- Reduce across the 16 lanes of each half (columns of each row) ----
#pragma unroll
  for (int v = 0; v < 8; ++v) {
#pragma unroll
    for (int off = 1; off < 16; off <<= 1) {
      rsum[v] += __shfl_xor(rsum[v], off, 32);   // masks 1,2,4,8 stay within halves
      spos[v] += __shfl_xor(spos[v], off, 32);
    }
  }

  // Every lane of a half now holds the 8 row sums for its half's rows.
  float loss = 0.0f;
#pragma unroll
  for (int v = 0; v < 8; ++v) {
    const float denom = fmaxf(rsum[v], 1e-8f);
    const float p     = fmaxf(spos[v], 1e-8f);
    loss += __logf(denom) - __logf(p);           // = -(log p - log denom)
  }

  // One atomic per 16-lane half (lanes 0 and 16), pre-scaled by 1/N for the mean.
  if (l16 == 0)
    atomicAdd(out, loss * (1.0f / (float)N_ROWS));
}

extern "C" void kernel_launch(void* const* d_in, const int* in_sizes, int n_in,
                              void* d_out, int out_size, void* d_ws, size_t ws_size,
                              hipStream_t stream) {
  (void)in_sizes; (void)n_in; (void)out_size; (void)d_ws; (void)ws_size;
  const float* za = (const float*)d_in[0];
  const float* zb = (const float*)d_in[1];
  float* out = (float*)d_out;

  zero_out_kernel<<<dim3(1), dim3(1), 0, stream>>>(out);

  // 1024 M-tiles, 8 waves (M-tiles) per 256-thread block -> 128 blocks
  infonce_wmma_kernel<<<dim3(NTILES / 8), dim3(256), 0, stream>>>(za, zb, out);
}